// DFFN_19301583028759
// MI455X (gfx1250) — compile-verified
//
#include <hip/hip_runtime.h>
#include <math.h>

typedef float v2f __attribute__((ext_vector_type(2)));
typedef float v4f __attribute__((ext_vector_type(4)));
typedef float v8f __attribute__((ext_vector_type(8)));

#define C2    384
#define HID   192
#define CIN   96
#define BATCH 4
#define HWSZ  65536   // 256*256

// ---------------------------------------------------------------------------
// K0: spectral gate -> per-channel 8x8 circular-conv taps
// h_c[d1,d2] = (1/64) * sum_k W_c[k1,k2] * cos(2*pi*(k1*d1+k2*d2)/8)
// W_c is the Hermitian extension of fft_w[c] (shape [8][5]); the
// self-conjugate columns k2=0,4 are symmetrized (FFTW c2r ignores Im there).
// ---------------------------------------------------------------------------
__global__ void taps_kernel(const float* __restrict__ fftw,
                            float* __restrict__ taps) {
  int idx = blockIdx.x * blockDim.x + threadIdx.x;
  if (idx >= C2 * 64) return;
  int c  = idx >> 6;
  int d  = idx & 63;
  int d1 = d >> 3, d2 = d & 7;
  const float* wc = fftw + c * 40;   // [8][5]
  float s = 0.f;
  for (int k1 = 0; k1 < 8; ++k1) {
    for (int k2 = 0; k2 < 8; ++k2) {
      float wv;
      if (k2 == 0 || k2 == 4) {
        wv = 0.5f * (wc[k1 * 5 + k2] + wc[((8 - k1) & 7) * 5 + k2]);
      } else if (k2 < 4) {
        wv = wc[k1 * 5 + k2];
      } else {
        wv = wc[((8 - k1) & 7) * 5 + (8 - k2)];
      }
      int m = (k1 * d1 + k2 * d2) & 7;
      s += wv * cosf(0.78539816339744831f * (float)m);  // cos(2*pi*m/8)
    }
  }
  taps[idx] = s * (1.0f / 64.0f);
}

// ---------------------------------------------------------------------------
// K1: fused project_in (1x1 conv == GEMM) + spectral gate (circular conv).
// One block = one 8x8 patch x 192 channels (half of C2).
// x patch staged into LDS via GLOBAL_LOAD_ASYNC_TO_LDS_B128 (ASYNCcnt path).
// GEMM: [192ch x 96] x [96 x 64px] via V_WMMA_F32_16X16X4_F32.
// ---------------------------------------------------------------------------
__global__ __launch_bounds__(256)
void proj_in_gate_kernel(const float* __restrict__ x,
                         const float* __restrict__ w_in,
                         const float* __restrict__ taps,
                         float* __restrict__ t1) {
  __shared__ float lds[HID * 64];   // 48 KB; first 96*64 floats double as x-stage

  int blk   = blockIdx.x;
  int patch = blk >> 1;
  int mhalf = blk & 1;              // which 192-channel half
  int b  = patch >> 10;
  int ph = (patch >> 5) & 31;
  int pw = patch & 31;
  int tid  = threadIdx.x;
  int lane = tid & 31;
  int wv   = tid >> 5;
  int gy0 = ph << 3, gx0 = pw << 3;

  // ---- async-stage x patch into LDS: xs[k][p], 16B granules, 6 per thread.
  // Generic LDS pointer low 32 bits == workgroup-relative LDS byte offset.
  {
    unsigned int lds_base = (unsigned int)(size_t)(&lds[0]);
    const float* xb = x + ((size_t)(b * CIN) << 16);
    for (int i = 0; i < 6; ++i) {
      int f = (tid + (i << 8)) << 2;          // float index, multiple of 4
      int k = f >> 6, p = f & 63;
      int gy = gy0 + (p >> 3), gx = gx0 + (p & 7);
      const float* gp = xb + ((size_t)k << 16) + (gy << 8) + gx;
      unsigned int ldsoff = lds_base + ((unsigned int)f << 2);
      asm volatile("global_load_async_to_lds_b128 %0, %1, off"
                   :: "v"(ldsoff), "v"(gp) : "memory");
    }
    asm volatile("s_wait_asynccnt 0x0" ::: "memory");
  }
  __syncthreads();
  float* xs = lds;

  int ntile = wv & 3;
  int mbase = wv >> 2;                        // 0 or 1
  int koff  = (lane < 16) ? 0 : 2;            // A/B fragment K offset per ISA layout
  int nn    = (ntile << 4) + (lane & 15);     // pixel within patch

  v8f z = {0.f, 0.f, 0.f, 0.f, 0.f, 0.f, 0.f, 0.f};
  v8f acc[6] = {z, z, z, z, z, z};

  for (int k0 = 0; k0 < CIN; k0 += 4) {
    v2f bf;
    bf.x = xs[(k0 + koff) * 64 + nn];
    bf.y = xs[(k0 + koff + 1) * 64 + nn];
#pragma unroll
    for (int j = 0; j < 6; ++j) {
      int mt   = mbase + (j << 1);                          // local M-tile 0..11
      int rowg = mhalf * HID + (mt << 4) + (lane & 15);     // global channel
      v2f af = *(const v2f*)(w_in + rowg * CIN + k0 + koff);
      acc[j] = __builtin_amdgcn_wmma_f32_16x16x4_f32(
          false, af, false, bf, (short)0, acc[j], false, false);
    }
  }
  __syncthreads();   // everyone done reading xs before we overwrite lds with y

  // ---- scatter C fragments to LDS as y[c_local][p]
  int rhalf = (lane >> 4) << 3;   // M += 8 for lanes 16..31
#pragma unroll
  for (int j = 0; j < 6; ++j) {
    int mt = mbase + (j << 1);
#pragma unroll
    for (int v = 0; v < 8; ++v) {
      int cl = (mt << 4) + v + rhalf;
      lds[(cl << 6) + nn] = acc[j][v];
    }
  }
  __syncthreads();

  // ---- spectral gate: per-channel 8x8 circular conv, write to t1
  for (int o = tid; o < HID * 64; o += 256) {
    int cl = o >> 6, p = o & 63;
    int cg = mhalf * HID + cl;
    int p1 = p >> 3, p2 = p & 7;
    const float* tc = taps + (cg << 6);
    const float* yc = lds + (cl << 6);
    float s = 0.f;
#pragma unroll 8
    for (int q = 0; q < 64; ++q) {
      int dd = (((p1 - (q >> 3)) & 7) << 3) | ((p2 - (q & 7)) & 7);
      s = fmaf(tc[dd], yc[q], s);
    }
    t1[((b * C2 + cg) << 16) + ((gy0 + p1) << 8) + (gx0 + p2)] = s;
  }
}

// ---------------------------------------------------------------------------
// K2: 3x3 depthwise conv (SAME, zero pad) on both channel halves + exact-erf
// GELU gating: out = gelu(dw(x[:192])) * dw(x[192:]).  4 pixels/thread,
// 6-wide row reads reused across taps, 16B vector store.  Memory-bound:
// ~600 MB of traffic -> this kernel should be pure-bandwidth at 23.3 TB/s.
// ---------------------------------------------------------------------------
__global__ __launch_bounds__(256)
void dw_gelu_kernel(const float* __restrict__ t1,
                    const float* __restrict__ w_dw,
                    float* __restrict__ t2) {
  int idx = blockIdx.x * 256 + threadIdx.x;   // BATCH*HID*HWSZ/4 threads
  int x0 = (idx & 63) << 2;                   // 64 pixel-quads per row
  int y  = (idx >> 6) & 255;
  int t  = idx >> 14;
  int c  = t % HID;
  int b  = t / HID;
  const float* p1 = t1 + ((size_t)(b * C2 + c) << 16);
  const float* p2 = t1 + ((size_t)(b * C2 + c + HID) << 16);
  const float* wa = w_dw + c * 9;
  const float* wb = w_dw + (c + HID) * 9;
  float s1[4] = {0.f, 0.f, 0.f, 0.f};
  float s2[4] = {0.f, 0.f, 0.f, 0.f};
#pragma unroll
  for (int dy = -1; dy <= 1; ++dy) {
    int yy = y + dy;
    if (yy < 0 || yy > 255) continue;
    float r1[6], r2[6];
#pragma unroll
    for (int j = 0; j < 6; ++j) {
      int xc = x0 - 1 + j;
      bool ok = (xc >= 0) && (xc < 256);
      int off = (yy << 8) + (ok ? xc : 0);
      r1[j] = ok ? p1[off] : 0.f;
      r2[j] = ok ? p2[off] : 0.f;
    }
#pragma unroll
    for (int dx = 0; dx < 3; ++dx) {
      float wva = wa[(dy + 1) * 3 + dx];
      float wvb = wb[(dy + 1) * 3 + dx];
#pragma unroll
      for (int u = 0; u < 4; ++u) {
        s1[u] = fmaf(wva, r1[u + dx], s1[u]);
        s2[u] = fmaf(wvb, r2[u + dx], s2[u]);
      }
    }
  }
  v4f o;
#pragma unroll
  for (int u = 0; u < 4; ++u) {
    float g = 0.5f * s1[u] * (1.f + erff(s1[u] * 0.70710678118654752f));
    o[u] = g * s2[u];
  }
  *(v4f*)(t2 + ((size_t)(b * HID + c) << 16) + (y << 8) + x0) = o;
}

// ---------------------------------------------------------------------------
// K3: project_out GEMM  out[96 x N] = w_out[96x192] * t2[192 x N], N = B*HW.
// Each wave: one 16-pixel column block, all 6 M-tiles; K stepped by 4 (fp32 WMMA).
// ---------------------------------------------------------------------------
__global__ __launch_bounds__(256)
void proj_out_kernel(const float* __restrict__ t2,
                     const float* __restrict__ w_out,
                     float* __restrict__ out) {
  int tid  = threadIdx.x;
  int lane = tid & 31;
  int wv   = tid >> 5;
  int gw    = blockIdx.x * 8 + wv;        // global wave, 0..16383
  int b     = gw >> 12;                   // 4096 column blocks per batch image
  int chunk = gw & 4095;
  int px    = (chunk << 4) + (lane & 15);
  int koff  = (lane < 16) ? 0 : 2;

  v8f z = {0.f, 0.f, 0.f, 0.f, 0.f, 0.f, 0.f, 0.f};
  v8f acc[6] = {z, z, z, z, z, z};
  const float* tb = t2 + (size_t)((b * HID) << 16) + px;

  for (int k0 = 0; k0 < HID; k0 += 4) {
    v2f bf;
    bf.x = tb[(size_t)(k0 + koff) << 16];
    bf.y = tb[(size_t)(k0 + koff + 1) << 16];
#pragma unroll
    for (int m = 0; m < 6; ++m) {
      int row = (m << 4) + (lane & 15);
      v2f af = *(const v2f*)(w_out + row * HID + k0 + koff);
      acc[m] = __builtin_amdgcn_wmma_f32_16x16x4_f32(
          false, af, false, bf, (short)0, acc[m], false, false);
    }
  }

  int rhalf = (lane >> 4) << 3;
#pragma unroll
  for (int m = 0; m < 6; ++m) {
#pragma unroll
    for (int v = 0; v < 8; ++v) {
      int row = (m << 4) + v + rhalf;
      out[((b * CIN + row) << 16) + px] = acc[m][v];
    }
  }
}

// ---------------------------------------------------------------------------
extern "C" void kernel_launch(void* const* d_in, const int* in_sizes, int n_in,
                              void* d_out, int out_size, void* d_ws, size_t ws_size,
                              hipStream_t stream) {
  const float* x     = (const float*)d_in[0];
  const float* w_in  = (const float*)d_in[1];
  const float* w_dw  = (const float*)d_in[2];
  const float* fftw  = (const float*)d_in[3];
  const float* w_out = (const float*)d_in[4];
  float* out = (float*)d_out;

  // workspace layout (floats): taps[384*64] | t1[B*384*HW] | t2[B*192*HW]
  float* taps = (float*)d_ws;
  float* t1   = taps + (size_t)C2 * 64;
  float* t2   = t1 + (size_t)BATCH * C2 * HWSZ;

  taps_kernel<<<(C2 * 64 + 255) / 256, 256, 0, stream>>>(fftw, taps);
  proj_in_gate_kernel<<<BATCH * 32 * 32 * 2, 256, 0, stream>>>(x, w_in, taps, t1);
  dw_gelu_kernel<<<(BATCH * HID * HWSZ) / (4 * 256), 256, 0, stream>>>(t1, w_dw, t2);
  proj_out_kernel<<<2048, 256, 0, stream>>>(t2, w_out, out);
}